// SpGraphTransAttentionLayer_21122649162299
// MI455X (gfx1250) — compile-verified
//
#include <hip/hip_runtime.h>

#define N_NODES 100000
#define N_EDGES 3200000
#define D_IN    128
#define ATT_DIM 64
#define HEADS   4
#define D_K     16

typedef float v2f __attribute__((ext_vector_type(2)));
typedef float v8f __attribute__((ext_vector_type(8)));

// ---------------------------------------------------------------------------
// Monotonic float <-> uint mapping so float max can use integer atomicMax.
// ---------------------------------------------------------------------------
__device__ __forceinline__ unsigned fmap(float f) {
    unsigned u = __float_as_uint(f);
    return (u & 0x80000000u) ? ~u : (u | 0x80000000u);
}
__device__ __forceinline__ float funmap(unsigned u) {
    unsigned b = (u & 0x80000000u) ? (u & 0x7fffffffu) : ~u;
    return __uint_as_float(b);
}

// ---------------------------------------------------------------------------
// Kernel 0: init per-(node,head) running max (mapped) and sum buffers.
// mapped 0 is below the mapping of every finite float (incl. -inf).
// ---------------------------------------------------------------------------
__global__ void sgta_init_kernel(unsigned* __restrict__ mmax,
                                 float* __restrict__ ssum) {
    int i = blockIdx.x * blockDim.x + threadIdx.x;
    if (i < N_NODES * HEADS) {
        mmax[i] = 0u;
        ssum[i] = 0.0f;
    }
}

// ---------------------------------------------------------------------------
// Kernel 1: fused q/k/v projection GEMM using V_WMMA_F32_16X16X4_F32.
// One wave computes a 16-row tile of all three 64-wide projections.
// A (x tile) from global, B (weights) staged in LDS, f32 accumulation.
// v is written directly in the reference's transposed [N, d_k, heads] layout.
// ---------------------------------------------------------------------------
__global__ __launch_bounds__(256) void sgta_qkv_gemm_kernel(
    const float* __restrict__ x,
    const float* __restrict__ WQ, const float* __restrict__ bQ,
    const float* __restrict__ WK, const float* __restrict__ bK,
    const float* __restrict__ WV, const float* __restrict__ bV,
    float* __restrict__ qbuf, float* __restrict__ kbuf,
    float* __restrict__ vout)
{
    __shared__ float sW[3][D_IN * ATT_DIM];   // 3 x 32 KB
    __shared__ float sB[3][ATT_DIM];

    {   // cooperative stage of weights (float4) + biases into LDS
        const float* wsrc[3] = {WQ, WK, WV};
        const int q4 = D_IN * ATT_DIM / 4;    // 2048 float4 per matrix
        for (int i = threadIdx.x; i < 3 * q4; i += blockDim.x) {
            int m = i / q4, off = i - m * q4;
            ((float4*)sW[m])[off] = ((const float4*)wsrc[m])[off];
        }
        const float* bsrc[3] = {bQ, bK, bV};
        if (threadIdx.x < 3 * ATT_DIM) {
            int m = threadIdx.x / ATT_DIM, c = threadIdx.x - m * ATT_DIM;
            sB[m][c] = bsrc[m][c];
        }
    }
    __syncthreads();

    const int wave   = threadIdx.x >> 5;
    const int lane   = threadIdx.x & 31;
    const int laneLo = lane & 15;
    const int laneHi = lane >> 4;
    const int tile   = blockIdx.x * 8 + wave;
    if (tile >= N_NODES / 16) return;         // whole-wave guard: EXEC all-1s
    const int row0 = tile * 16;

    v8f c[12];                                 // [matrix 0..2][colTile 0..3]
    v8f zero = {0.f,0.f,0.f,0.f,0.f,0.f,0.f,0.f};
#pragma unroll
    for (int t = 0; t < 12; ++t) c[t] = zero;

    // A-fragment base: lane<16 holds K pairs {4j,4j+1}, lane>=16 {4j+2,4j+3}
    const float* xrow = x + (size_t)(row0 + laneLo) * D_IN + 2 * laneHi;

    for (int j = 0; j < D_IN / 4; ++j) {      // 32 K-steps of 4
        v2f a = *(const v2f*)(xrow + 4 * j);  // aligned 8B load
        const int kb = 4 * j + 2 * laneHi;    // K row held in b.x
#pragma unroll
        for (int m = 0; m < 3; ++m) {
#pragma unroll
            for (int ct = 0; ct < 4; ++ct) {
                v2f b;
                b.x = sW[m][(kb + 0) * ATT_DIM + ct * 16 + laneLo];
                b.y = sW[m][(kb + 1) * ATT_DIM + ct * 16 + laneLo];
                c[m * 4 + ct] = __builtin_amdgcn_wmma_f32_16x16x4_f32(
                    false, a, false, b, (short)0, c[m * 4 + ct], false, false);
            }
        }
    }

    // D layout: VGPR r -> M = r + 8*laneHi, N = laneLo
#pragma unroll
    for (int ct = 0; ct < 4; ++ct) {
#pragma unroll
        for (int r = 0; r < 8; ++r) {
            const int row = row0 + r + 8 * laneHi;
            const int col = ct * 16 + laneLo;           // head = ct, d = laneLo
            float qv = c[0 * 4 + ct][r] + sB[0][col];
            float kv = c[1 * 4 + ct][r] + sB[1][col];
            float vv = c[2 * 4 + ct][r] + sB[2][col];
            qbuf[(size_t)row * ATT_DIM + col] = qv;
            kbuf[(size_t)row * ATT_DIM + col] = kv;
            // v_out[n, d, h] = v[n, h, d]
            vout[(size_t)row * ATT_DIM + laneLo * HEADS + ct] = vv;
        }
    }
}

// ---------------------------------------------------------------------------
// Kernel 2: per-edge scaled dot products (4 heads) + atomic segment max.
// q/k fit in L2 (25.6 MB each) so gathers stay on-chip.
// ---------------------------------------------------------------------------
__global__ __launch_bounds__(256) void sgta_edge_dot_kernel(
    const int* __restrict__ e0, const int* __restrict__ e1,
    const float* __restrict__ q, const float* __restrict__ k,
    float* __restrict__ prods, unsigned* __restrict__ mmax)
{
    int e = blockIdx.x * blockDim.x + threadIdx.x;
    if (e >= N_EDGES) return;
    const int s = e0[e];
    const int d = e1[e];
    const float4* qp = (const float4*)(q + (size_t)s * ATT_DIM);
    const float4* kp = (const float4*)(k + (size_t)d * ATT_DIM);
#pragma unroll
    for (int h = 0; h < HEADS; ++h) {
        float dot = 0.0f;
#pragma unroll
        for (int i = 0; i < D_K / 4; ++i) {
            float4 a = qp[h * 4 + i];
            float4 b = kp[h * 4 + i];
            dot += a.x * b.x + a.y * b.y + a.z * b.z + a.w * b.w;
        }
        dot *= 0.25f;                          // 1/sqrt(d_k), d_k = 16
        prods[(size_t)e * HEADS + h] = dot;
        atomicMax(&mmax[s * HEADS + h], fmap(dot));
    }
}

// ---------------------------------------------------------------------------
// Kernel 3: ex = exp(prod - segmax); stash ex in attention slot; atomic sum.
// ---------------------------------------------------------------------------
__global__ __launch_bounds__(256) void sgta_edge_exp_kernel(
    const int* __restrict__ e0,
    const float* __restrict__ prods,
    const unsigned* __restrict__ mmax,
    float* __restrict__ att, float* __restrict__ ssum)
{
    int e = blockIdx.x * blockDim.x + threadIdx.x;
    if (e >= N_EDGES) return;
    const int s = e0[e];
#pragma unroll
    for (int h = 0; h < HEADS; ++h) {
        float mv = funmap(mmax[s * HEADS + h]);
        float ex = expf(prods[(size_t)e * HEADS + h] - mv);
        att[(size_t)e * HEADS + h] = ex;
        atomicAdd(&ssum[s * HEADS + h], ex);
    }
}

// ---------------------------------------------------------------------------
// Kernel 4: normalize: attention = ex / (segsum + 1e-16)
// ---------------------------------------------------------------------------
__global__ __launch_bounds__(256) void sgta_edge_norm_kernel(
    const int* __restrict__ e0,
    const float* __restrict__ ssum,
    float* __restrict__ att)
{
    int e = blockIdx.x * blockDim.x + threadIdx.x;
    if (e >= N_EDGES) return;
    const int s = e0[e];
#pragma unroll
    for (int h = 0; h < HEADS; ++h) {
        float den = ssum[s * HEADS + h] + 1e-16f;
        att[(size_t)e * HEADS + h] = att[(size_t)e * HEADS + h] / den;
    }
}

// ---------------------------------------------------------------------------
extern "C" void kernel_launch(void* const* d_in, const int* in_sizes, int n_in,
                              void* d_out, int out_size, void* d_ws, size_t ws_size,
                              hipStream_t stream) {
    const float* x  = (const float*)d_in[0];
    const int*   eg = (const int*)d_in[1];        // edge[2, E] int32
    const float* WQ = (const float*)d_in[2];
    const float* bQ = (const float*)d_in[3];
    const float* WK = (const float*)d_in[4];
    const float* bK = (const float*)d_in[5];
    const float* WV = (const float*)d_in[6];
    const float* bV = (const float*)d_in[7];

    const int* e0 = eg;
    const int* e1 = eg + N_EDGES;

    // d_out = [attention (E*H) | v_out (N*64) | prods (E*H)]
    float* out   = (float*)d_out;
    float* att   = out;
    float* vout  = att + (size_t)N_EDGES * HEADS;
    float* prods = vout + (size_t)N_NODES * ATT_DIM;

    // workspace: q (N*64) | k (N*64) | mmax (N*H u32) | ssum (N*H f32)
    float*    qbuf = (float*)d_ws;
    float*    kbuf = qbuf + (size_t)N_NODES * ATT_DIM;
    unsigned* mmax = (unsigned*)(kbuf + (size_t)N_NODES * ATT_DIM);
    float*    ssum = (float*)(mmax + (size_t)N_NODES * HEADS);

    const int nh_blocks   = (N_NODES * HEADS + 255) / 256;
    const int gemm_blocks = (N_NODES / 16 + 7) / 8;          // 8 waves/block
    const int edge_blocks = (N_EDGES + 255) / 256;

    sgta_init_kernel<<<nh_blocks, 256, 0, stream>>>(mmax, ssum);
    sgta_qkv_gemm_kernel<<<gemm_blocks, 256, 0, stream>>>(
        x, WQ, bQ, WK, bK, WV, bV, qbuf, kbuf, vout);
    sgta_edge_dot_kernel<<<edge_blocks, 256, 0, stream>>>(
        e0, e1, qbuf, kbuf, prods, mmax);
    sgta_edge_exp_kernel<<<edge_blocks, 256, 0, stream>>>(
        e0, prods, mmax, att, ssum);
    sgta_edge_norm_kernel<<<edge_blocks, 256, 0, stream>>>(e0, ssum, att);
}